// ConvLRU_49855980372272
// MI455X (gfx1250) — compile-verified
//
#include <hip/hip_runtime.h>
#include <stdint.h>

// ============================================================================
// ConvLRU for MI455X (gfx1250, wave32). All GEMM-shaped stages run on
// V_WMMA_F32_16X16X4_F32; GEMM panel staging uses the CDNA5 async
// global->LDS path (GLOBAL_LOAD_ASYNC_TO_LDS_B128 + s_wait_asynccnt).
// Pipeline (two complex ping-pong buffers A,B in ws; both fit in 192MB L2):
//   init tabs -> rowDFT(x@F)->B -> colDFT(F@B)->A -> zin(A,U*,V*) ->
//   zscan -> hscan(A in place) -> expand(A += (U.z)@V^T) ->
//   chanmix(pw^T@A)->B -> inv colDFT->A -> inv rowDFT->B ->
//   conv3x3(B.re)->A.re, conv3x3(B.im)->A.im -> proj(fw)->B.re ->
//   layernorm+residual -> d_out
// Workspace need: ~135 MB (assumed available in d_ws).
// ============================================================================

typedef __attribute__((ext_vector_type(2))) float v2f;
typedef __attribute__((ext_vector_type(8))) float v8f;

// D = A(16x4) * B(4x16) + C, fp32 WMMA.
__device__ inline v8f wmma4(v2f a, v2f b, v8f c) {
  return __builtin_amdgcn_wmma_f32_16x16x4_f32(
      false, a, false, b, (short)0, c, false, false);
}

// Complex MAC on a 16x16 tile: (cr + i ci) += (ar + i ai)(br + i bi).
// fp32 WMMA has no A/B NEG modifier (only CNeg), so negate the Ai fragment.
__device__ inline void cmac(v8f& cr, v8f& ci, v2f ar, v2f ai, v2f br, v2f bi) {
  v2f ain; ain.x = -ai.x; ain.y = -ai.y;
  cr = wmma4(ar, br, cr);
  cr = wmma4(ain, bi, cr);
  ci = wmma4(ar, bi, ci);
  ci = wmma4(ai, br, ci);
}

// CDNA5 async global->LDS copy, 16 bytes per lane. The LDS destination
// operand wants the raw LDS byte offset, which is the low 32 bits of the
// generic address (aperture scheme, ISA 10.2). Tracked by ASYNCcnt.
__device__ inline void async_copy_b128(const float* gsrc, float* lds_dst) {
  unsigned lds = (unsigned)(uintptr_t)lds_dst;
  unsigned long long ga = (unsigned long long)(uintptr_t)gsrc;
  asm volatile("global_load_async_to_lds_b128 %0, %1, off"
               :: "v"(lds), "v"(ga) : "memory");
}
__device__ inline void wait_async_all() {
  asm volatile("s_wait_asynccnt 0x0" ::: "memory");
}

// ---------------------------------------------------------------------------
// Init: DFT tables (fwd: e^{-i2pi jk/128}; inv: e^{+i...}/128) and LRU params.
// ---------------------------------------------------------------------------
__global__ __launch_bounds__(256) void clru_init(
    float* __restrict__ TFre, float* __restrict__ TFim,
    float* __restrict__ TIre, float* __restrict__ TIim,
    const float* __restrict__ nu_s, const float* __restrict__ th_s,
    const float* __restrict__ gm_s,
    const float* __restrict__ nu_r, const float* __restrict__ th_r,
    const float* __restrict__ gm_r,
    float* __restrict__ lamSre, float* __restrict__ lamSim,
    float* __restrict__ gamS,
    float* __restrict__ lamRre, float* __restrict__ lamRim,
    float* __restrict__ gamR) {
  int idx = blockIdx.x * 256 + threadIdx.x;
  if (idx < 16384) {
    int j = idx >> 7, k = idx & 127;
    int m = (j * k) & 127;  // exact phase mod N for accuracy
    float ang = 6.283185307179586f * (float)m * (1.0f / 128.0f);
    float c = cosf(ang), s = sinf(ang);
    TFre[idx] = c;               TFim[idx] = -s;
    TIre[idx] = c * (1.f/128.f); TIim[idx] = s * (1.f/128.f);
  }
  if (idx < 4096) {  // lam_s over (C,S)
    float mod = expf(-expf(nu_s[idx]));
    float a = expf(th_s[idx]);
    lamSre[idx] = mod * cosf(a);
    lamSim[idx] = mod * sinf(a);
    gamS[idx] = expf(gm_s[idx]);
  }
  if (idx < 1024) {  // lam_r over (C,R)
    float mod = expf(-expf(nu_r[idx]));
    float a = expf(th_r[idx]);
    lamRre[idx] = mod * cosf(a);
    lamRim[idx] = mod * sinf(a);
    gamR[idx] = expf(gm_r[idx]);
  }
}

// ---------------------------------------------------------------------------
// Generic 128x128x128 complex GEMM per image: out[n] = P[n?] @ Q[n?].
// The per-image operand has stride 16384; the shared DFT table has stride 0.
// PREAL (compile-time): P is real (forward row-DFT of real x).
// 8 waves; each wave owns a 16-row strip x 8 col tiles; K-panels are staged
// into LDS with GLOBAL_LOAD_ASYNC_TO_LDS_B128 (16B-aligned chunks).
// ---------------------------------------------------------------------------
template <bool PREAL>
__global__ __launch_bounds__(256) void clru_cgemm128(
    float* __restrict__ outRe, float* __restrict__ outIm, long outStride,
    const float* __restrict__ pRe, const float* __restrict__ pIm, long pStride,
    const float* __restrict__ qRe, const float* __restrict__ qIm, long qStride) {
  __shared__ float Pre[128 * 16], Pim[128 * 16];
  __shared__ float Qre[16 * 128], Qim[16 * 128];
  const int n = blockIdx.x;
  const int tid = threadIdx.x;
  const int lane = tid & 31, wv = tid >> 5;
  const int mm = lane & 15, hi = lane >> 4;
  const float* pR = pRe + (long)n * pStride;
  const float* pI = PREAL ? nullptr : (pIm + (long)n * pStride);
  const float* qR = qRe + (long)n * qStride;
  const float* qI = qIm + (long)n * qStride;

  v8f cre[8] = {}; v8f cim[8] = {};
  for (int kb = 0; kb < 8; ++kb) {
    __syncthreads();  // previous iteration's readers are done
    // Async-stage panels (512 x 16B chunks per array; 2 per thread).
    for (int q = tid; q < 512; q += 256) {
      int r = q >> 2, sub4 = (q & 3) * 4;     // P: 16-float rows
      async_copy_b128(pR + r * 128 + kb * 16 + sub4, &Pre[r * 16 + sub4]);
      if (!PREAL)
        async_copy_b128(pI + r * 128 + kb * 16 + sub4, &Pim[r * 16 + sub4]);
      int k2 = q >> 3, c16 = (q & 7) * 16;    // Q: 128-float rows
      async_copy_b128(qR + (kb * 16 + k2) * 128 + c16, &Qre[k2 * 128 + c16]);
      async_copy_b128(qI + (kb * 16 + k2) * 128 + c16, &Qim[k2 * 128 + c16]);
    }
    wait_async_all();  // s_wait_asynccnt 0: this wave's LDS writes landed
    __syncthreads();   // all waves' panels visible
#pragma unroll
    for (int k4 = 0; k4 < 4; ++k4) {
      const int k0 = k4 * 4;
      v2f ar, ai;
      ar.x = Pre[(wv * 16 + mm) * 16 + k0 + 2 * hi];
      ar.y = Pre[(wv * 16 + mm) * 16 + k0 + 2 * hi + 1];
      if (!PREAL) {
        ai.x = Pim[(wv * 16 + mm) * 16 + k0 + 2 * hi];
        ai.y = Pim[(wv * 16 + mm) * 16 + k0 + 2 * hi + 1];
      }
#pragma unroll
      for (int t = 0; t < 8; ++t) {
        v2f br, bi;
        br.x = Qre[(k0 + 2 * hi) * 128 + t * 16 + mm];
        br.y = Qre[(k0 + 2 * hi + 1) * 128 + t * 16 + mm];
        bi.x = Qim[(k0 + 2 * hi) * 128 + t * 16 + mm];
        bi.y = Qim[(k0 + 2 * hi + 1) * 128 + t * 16 + mm];
        if (PREAL) {
          cre[t] = wmma4(ar, br, cre[t]);
          cim[t] = wmma4(ar, bi, cim[t]);
        } else {
          cmac(cre[t], cim[t], ar, ai, br, bi);
        }
      }
    }
  }
  float* oR = outRe + (long)n * outStride;
  float* oI = outIm + (long)n * outStride;
#pragma unroll
  for (int t = 0; t < 8; ++t)
#pragma unroll
    for (int i = 0; i < 8; ++i) {
      int row = wv * 16 + i + 8 * hi, col = t * 16 + mm;
      oR[row * 128 + col] = cre[t][i];
      oI[row * 128 + col] = cim[t][i];
    }
}

// ---------------------------------------------------------------------------
// z_in[l,c,r] = sum_s conj(U[c,s,r]) * (Xc[l,c] @ conj(V[c]))[s,r]
// Phase 1: P(128x32) via WMMA; Phase 2: conj-U weighted reduction over s.
// ---------------------------------------------------------------------------
__global__ __launch_bounds__(256) void clru_zin(
    const float* __restrict__ XR, const float* __restrict__ XI,
    const float* __restrict__ Ure, const float* __restrict__ Uim,
    const float* __restrict__ Vre, const float* __restrict__ Vim,
    float* __restrict__ zinRe, float* __restrict__ zinIm) {
  __shared__ float Pre[128 * 32], Pim[128 * 32];
  __shared__ float redR[256], redI[256];
  const int lc = blockIdx.x;      // l*C + c
  const int c = lc & 31;
  const int tid = threadIdx.x;
  const int lane = tid & 31, wv = tid >> 5;
  const int mm = lane & 15, hi = lane >> 4;
  const float* xr = XR + (long)lc * 16384;
  const float* xi = XI + (long)lc * 16384;

  v8f pr[2] = {}; v8f pi[2] = {};
  for (int k = 0; k < 128; k += 4) {
    v2f ar, ai;
    ar.x = xr[(wv * 16 + mm) * 128 + k + 2 * hi];
    ar.y = xr[(wv * 16 + mm) * 128 + k + 2 * hi + 1];
    ai.x = xi[(wv * 16 + mm) * 128 + k + 2 * hi];
    ai.y = xi[(wv * 16 + mm) * 128 + k + 2 * hi + 1];
#pragma unroll
    for (int t = 0; t < 2; ++t) {  // Q = conj(V[c]) : W x R
      v2f br, bi;
      br.x =  Vre[c * 4096 + (k + 2 * hi) * 32 + t * 16 + mm];
      br.y =  Vre[c * 4096 + (k + 2 * hi + 1) * 32 + t * 16 + mm];
      bi.x = -Vim[c * 4096 + (k + 2 * hi) * 32 + t * 16 + mm];
      bi.y = -Vim[c * 4096 + (k + 2 * hi + 1) * 32 + t * 16 + mm];
      cmac(pr[t], pi[t], ar, ai, br, bi);
    }
  }
#pragma unroll
  for (int t = 0; t < 2; ++t)
#pragma unroll
    for (int i = 0; i < 8; ++i) {
      int row = wv * 16 + i + 8 * hi, col = t * 16 + mm;
      Pre[row * 32 + col] = pr[t][i];
      Pim[row * 32 + col] = pi[t][i];
    }
  __syncthreads();
  // reduce over s with conj(U)
  const int r = tid & 31, chunk = tid >> 5;
  float sr = 0.f, si = 0.f;
  for (int s = chunk * 16; s < chunk * 16 + 16; ++s) {
    float ur = Ure[c * 4096 + s * 32 + r];
    float ui = Uim[c * 4096 + s * 32 + r];
    float p_r = Pre[s * 32 + r], p_i = Pim[s * 32 + r];
    sr += ur * p_r + ui * p_i;  // Re(conj(U)*P)
    si += ur * p_i - ui * p_r;  // Im(conj(U)*P)
  }
  redR[tid] = sr; redI[tid] = si;
  __syncthreads();
  if (tid < 32) {
    float ar = 0.f, ai = 0.f;
    for (int ch = 0; ch < 8; ++ch) { ar += redR[ch * 32 + tid]; ai += redI[ch * 32 + tid]; }
    zinRe[(long)lc * 32 + tid] = ar;
    zinIm[(long)lc * 32 + tid] = ai;
  }
}

// ---------------------------------------------------------------------------
// z recurrence over L (tiny): z = lam_r*z + gam_r*z_in, store zs[l].
// ---------------------------------------------------------------------------
__global__ void clru_zscan(
    const float* __restrict__ zinRe, const float* __restrict__ zinIm,
    const float* __restrict__ lamRre, const float* __restrict__ lamRim,
    const float* __restrict__ gamR,
    float* __restrict__ zsRe, float* __restrict__ zsIm) {
  int t = blockIdx.x * blockDim.x + threadIdx.x;  // (c,r) in C*R = 1024
  if (t >= 1024) return;
  float lr = lamRre[t], li = lamRim[t], g = gamR[t];
  float zr = 0.f, zi = 0.f;
  for (int l = 0; l < 16; ++l) {
    long idx = (long)l * 1024 + t;
    float xr2 = zinRe[idx], xi2 = zinIm[idx];
    float nr = lr * zr - li * zi + g * xr2;
    float ni = lr * zi + li * zr + g * xi2;
    zr = nr; zi = ni;
    zsRe[idx] = zr; zsIm[idx] = zi;
  }
}

// ---------------------------------------------------------------------------
// h recurrence over L, in place on the spectrum buffer (xc_l dead at step l).
// ---------------------------------------------------------------------------
__global__ __launch_bounds__(256) void clru_hscan(
    float* __restrict__ XR, float* __restrict__ XI,
    const float* __restrict__ lamSre, const float* __restrict__ lamSim,
    const float* __restrict__ gamS) {
  long t = (long)blockIdx.x * 256 + threadIdx.x;  // < C*S*W = 524288
  int c = (int)(t >> 14);
  int s = (int)((t >> 7) & 127);
  float lr = lamSre[c * 128 + s], li = lamSim[c * 128 + s], g = gamS[c * 128 + s];
  float hr = 0.f, hh = 0.f;
  for (int l = 0; l < 16; ++l) {
    long idx = (long)l * 524288 + t;
    float xr2 = XR[idx], xi2 = XI[idx];
    float nr = lr * hr - li * hh + g * xr2;
    float ni = lr * hh + li * hr + g * xi2;
    hr = nr; hh = ni;
    XR[idx] = hr; XI[idx] = hh;
  }
}

// ---------------------------------------------------------------------------
// Low-rank expansion: A[l,c] += (U[c] . zs[l,c]) @ V[c]^T  (128x128x32 cGEMM)
// A2 and V^T staged in LDS (64 KB).
// ---------------------------------------------------------------------------
__global__ __launch_bounds__(256) void clru_expand(
    float* __restrict__ XR, float* __restrict__ XI,
    const float* __restrict__ Ure, const float* __restrict__ Uim,
    const float* __restrict__ Vre, const float* __restrict__ Vim,
    const float* __restrict__ zsRe, const float* __restrict__ zsIm) {
  __shared__ float A2r[128 * 32], A2i[128 * 32];
  __shared__ float Qr[32 * 128], Qi[32 * 128];
  const int lc = blockIdx.x;
  const int c = lc & 31;
  const int tid = threadIdx.x;
  const int lane = tid & 31, wv = tid >> 5;
  const int mm = lane & 15, hi = lane >> 4;
  for (int idx = tid; idx < 4096; idx += 256) {
    int r = idx & 31;
    float zr = zsRe[(long)lc * 32 + r], zi = zsIm[(long)lc * 32 + r];
    float ur = Ure[c * 4096 + idx], ui = Uim[c * 4096 + idx];  // idx = s*32+r
    A2r[idx] = ur * zr - ui * zi;
    A2i[idx] = ur * zi + ui * zr;
    int w = idx >> 5;                       // idx = w*32+r for V
    Qr[r * 128 + w] = Vre[c * 4096 + idx];  // Q[r][w] = V[c][w][r]
    Qi[r * 128 + w] = Vim[c * 4096 + idx];
  }
  __syncthreads();
  float* xr = XR + (long)lc * 16384;
  float* xi = XI + (long)lc * 16384;
  v8f cr[8], ci[8];
#pragma unroll
  for (int t = 0; t < 8; ++t)
#pragma unroll
    for (int i = 0; i < 8; ++i) {  // accumulate onto existing hs
      int row = wv * 16 + i + 8 * hi, col = t * 16 + mm;
      cr[t][i] = xr[row * 128 + col];
      ci[t][i] = xi[row * 128 + col];
    }
#pragma unroll
  for (int k = 0; k < 32; k += 4) {
    v2f ar, ai;
    ar.x = A2r[(wv * 16 + mm) * 32 + k + 2 * hi];
    ar.y = A2r[(wv * 16 + mm) * 32 + k + 2 * hi + 1];
    ai.x = A2i[(wv * 16 + mm) * 32 + k + 2 * hi];
    ai.y = A2i[(wv * 16 + mm) * 32 + k + 2 * hi + 1];
#pragma unroll
    for (int t = 0; t < 8; ++t) {
      v2f br, bi;
      br.x = Qr[(k + 2 * hi) * 128 + t * 16 + mm];
      br.y = Qr[(k + 2 * hi + 1) * 128 + t * 16 + mm];
      bi.x = Qi[(k + 2 * hi) * 128 + t * 16 + mm];
      bi.y = Qi[(k + 2 * hi + 1) * 128 + t * 16 + mm];
      cmac(cr[t], ci[t], ar, ai, br, bi);
    }
  }
#pragma unroll
  for (int t = 0; t < 8; ++t)
#pragma unroll
    for (int i = 0; i < 8; ++i) {
      int row = wv * 16 + i + 8 * hi, col = t * 16 + mm;
      xr[row * 128 + col] = cr[t][i];
      xi[row * 128 + col] = ci[t][i];
    }
}

// ---------------------------------------------------------------------------
// Channel mix: Y[l,d,p] = sum_c pw[c,d] * H[l,c,p] + pb[d]   (complex)
// M=32 (2 tiles), K=32, N streamed 128 px/block.
// ---------------------------------------------------------------------------
__global__ __launch_bounds__(256) void clru_chanmix(
    const float* __restrict__ HR, const float* __restrict__ HI,
    const float* __restrict__ pwRe, const float* __restrict__ pwIm,
    const float* __restrict__ pbRe, const float* __restrict__ pbIm,
    float* __restrict__ YR, float* __restrict__ YI) {
  const int l = blockIdx.y;
  const int tid = threadIdx.x;
  const int lane = tid & 31, wv = tid >> 5;
  const int mm = lane & 15, hi = lane >> 4;
  const long px = (long)blockIdx.x * 128 + wv * 16;
  const float* hr = HR + (long)l * 524288;
  const float* hh = HI + (long)l * 524288;
  v8f cr[2] = {}; v8f ci[2] = {};
#pragma unroll
  for (int k = 0; k < 32; k += 4) {
    v2f br, bi;
    br.x = hr[(long)(k + 2 * hi) * 16384 + px + mm];
    br.y = hr[(long)(k + 2 * hi + 1) * 16384 + px + mm];
    bi.x = hh[(long)(k + 2 * hi) * 16384 + px + mm];
    bi.y = hh[(long)(k + 2 * hi + 1) * 16384 + px + mm];
#pragma unroll
    for (int dt = 0; dt < 2; ++dt) {  // A[m=d][k=c] = pw[c*32+d]
      v2f ar, ai;
      ar.x = pwRe[(k + 2 * hi) * 32 + dt * 16 + mm];
      ar.y = pwRe[(k + 2 * hi + 1) * 32 + dt * 16 + mm];
      ai.x = pwIm[(k + 2 * hi) * 32 + dt * 16 + mm];
      ai.y = pwIm[(k + 2 * hi + 1) * 32 + dt * 16 + mm];
      cmac(cr[dt], ci[dt], ar, ai, br, bi);
    }
  }
#pragma unroll
  for (int dt = 0; dt < 2; ++dt)
#pragma unroll
    for (int i = 0; i < 8; ++i) {
      int d = dt * 16 + i + 8 * hi;
      YR[(long)l * 524288 + (long)d * 16384 + px + mm] = cr[dt][i] + pbRe[d];
      YI[(long)l * 524288 + (long)d * 16384 + px + mm] = ci[dt][i] + pbIm[d];
    }
}

// ---------------------------------------------------------------------------
// 3x3 conv (pad 1) as implicit GEMM: M=32 co, K=288 (=2 ci-groups of 144),
// tile 16x32 output px/block; input halo tile staged in 38 KB LDS.
// ---------------------------------------------------------------------------
__global__ __launch_bounds__(256) void clru_conv3x3(
    const float* __restrict__ IN, const float* __restrict__ Wc,
    const float* __restrict__ Bc, float* __restrict__ OUT) {
  __shared__ float Ipan[16 * 18 * 34];
  const int c0 = blockIdx.x * 32;
  const int r0 = blockIdx.y * 16;
  const int n = blockIdx.z;
  const int tid = threadIdx.x;
  const int lane = tid & 31, wv = tid >> 5;
  const int mm = lane & 15, hi = lane >> 4;
  v8f acc[2][4] = {};
  for (int g = 0; g < 2; ++g) {
    __syncthreads();
    for (int idx = tid; idx < 16 * 18 * 34; idx += 256) {
      int ci = idx / 612, rem = idx - ci * 612;
      int rr = rem / 34, cc = rem - rr * 34;
      int gr = r0 - 1 + rr, gc = c0 - 1 + cc;
      float v = 0.f;
      if (gr >= 0 && gr < 128 && gc >= 0 && gc < 128)
        v = IN[((long)n * 32 + g * 16 + ci) * 16384 + gr * 128 + gc];
      Ipan[idx] = v;
    }
    if (g == 0)  // pull next ci-group toward the caches (global_prefetch_b8)
      __builtin_prefetch(
          &IN[((long)n * 32 + 16 + (tid & 15)) * 16384 + r0 * 128 + c0], 0, 1);
    __syncthreads();
    for (int kk = 0; kk < 144; kk += 4) {
      v2f aA[2];
#pragma unroll
      for (int ct = 0; ct < 2; ++ct) {  // A[m=co][k] = cw flat (co*288 + k)
        aA[ct].x = Wc[(ct * 16 + mm) * 288 + g * 144 + kk + 2 * hi];
        aA[ct].y = Wc[(ct * 16 + mm) * 288 + g * 144 + kk + 2 * hi + 1];
      }
      v2f bB[4];
#pragma unroll
      for (int nt = 0; nt < 4; ++nt) {
        int px = wv * 64 + nt * 16 + mm;
        int ph = px >> 5, pwc = px & 31;
        int ka = kk + 2 * hi;
        int cia = ka / 9, rema = ka - cia * 9;
        int kha = rema / 3, kwa = rema - kha * 3;
        bB[nt].x = Ipan[cia * 612 + (ph + kha) * 34 + (pwc + kwa)];
        int kb2 = ka + 1;
        int cib = kb2 / 9, remb = kb2 - cib * 9;
        int khb = remb / 3, kwb = remb - khb * 3;
        bB[nt].y = Ipan[cib * 612 + (ph + khb) * 34 + (pwc + kwb)];
      }
#pragma unroll
      for (int ct = 0; ct < 2; ++ct)
#pragma unroll
        for (int nt = 0; nt < 4; ++nt)
          acc[ct][nt] = wmma4(aA[ct], bB[nt], acc[ct][nt]);
    }
  }
#pragma unroll
  for (int ct = 0; ct < 2; ++ct)
#pragma unroll
    for (int nt = 0; nt < 4; ++nt)
#pragma unroll
      for (int i = 0; i < 8; ++i) {
        int co = ct * 16 + i + 8 * hi;
        int px = wv * 64 + nt * 16 + mm;
        int ph = px >> 5, pwc = px & 31;
        OUT[((long)n * 32 + co) * 16384 + (r0 + ph) * 128 + (c0 + pwc)] =
            acc[ct][nt][i] + Bc[co];
      }
}

// ---------------------------------------------------------------------------
// Final projection: proj[n,c,p] = sum_{k<64} fw[c,k]*cat[k,p] + fb[c],
// cat = [convr(32); convi(32)]. M=32 (2 tiles), K=64, N=128 px/block.
// ---------------------------------------------------------------------------
__global__ __launch_bounds__(256) void clru_proj(
    const float* __restrict__ CR, const float* __restrict__ CI,
    const float* __restrict__ fw, const float* __restrict__ fb,
    float* __restrict__ OUT) {
  const int n = blockIdx.y;
  const int tid = threadIdx.x;
  const int lane = tid & 31, wv = tid >> 5;
  const int mm = lane & 15, hi = lane >> 4;
  const long px = (long)blockIdx.x * 128 + wv * 16;
  v8f acc[2] = {};
#pragma unroll
  for (int k = 0; k < 64; k += 4) {
    int ka = k + 2 * hi, kb2 = ka + 1;  // (even,odd) pair stays in one half
    const float* sA = (ka < 32) ? CR + ((long)n * 32 + ka) * 16384
                                : CI + ((long)n * 32 + ka - 32) * 16384;
    const float* sB = (kb2 < 32) ? CR + ((long)n * 32 + kb2) * 16384
                                 : CI + ((long)n * 32 + kb2 - 32) * 16384;
    v2f b; b.x = sA[px + mm]; b.y = sB[px + mm];
#pragma unroll
    for (int ct = 0; ct < 2; ++ct) {
      v2f a;
      a.x = fw[(ct * 16 + mm) * 64 + ka];
      a.y = fw[(ct * 16 + mm) * 64 + kb2];
      acc[ct] = wmma4(a, b, acc[ct]);
    }
  }
#pragma unroll
  for (int ct = 0; ct < 2; ++ct)
#pragma unroll
    for (int i = 0; i < 8; ++i) {
      int c = ct * 16 + i + 8 * hi;
      OUT[((long)n * 32 + c) * 16384 + px + mm] = acc[ct][i] + fb[c];
    }
}

// ---------------------------------------------------------------------------
// LayerNorm over (S,W) per (l,c) + residual with x.
// ---------------------------------------------------------------------------
__global__ __launch_bounds__(256) void clru_ln_res(
    const float* __restrict__ P, const float* __restrict__ X,
    const float* __restrict__ g, const float* __restrict__ b2,
    float* __restrict__ OUT) {
  __shared__ float s1[256], s2[256];
  const long base = (long)blockIdx.x * 16384;
  const int tid = threadIdx.x;
  float sum = 0.f, sq = 0.f;
  for (int i = tid; i < 16384; i += 256) {
    float v = P[base + i];
    sum += v; sq += v * v;
  }
  s1[tid] = sum; s2[tid] = sq;
  __syncthreads();
  for (int off = 128; off > 0; off >>= 1) {
    if (tid < off) { s1[tid] += s1[tid + off]; s2[tid] += s2[tid + off]; }
    __syncthreads();
  }
  float mu = s1[0] * (1.f / 16384.f);
  float var = s2[0] * (1.f / 16384.f) - mu * mu;
  float rs = rsqrtf(var + 1e-5f);
  for (int i = tid; i < 16384; i += 256) {
    OUT[base + i] = (P[base + i] - mu) * rs * g[i] + b2[i] + X[base + i];
  }
}

// ============================================================================
// Host-side orchestration.
// ============================================================================
extern "C" void kernel_launch(void* const* d_in, const int* in_sizes, int n_in,
                              void* d_out, int out_size, void* d_ws,
                              size_t ws_size, hipStream_t stream) {
  (void)in_sizes; (void)n_in; (void)out_size; (void)ws_size;
  const float* x    = (const float*)d_in[0];
  const float* nu_s = (const float*)d_in[1];
  const float* th_s = (const float*)d_in[2];
  const float* gm_s = (const float*)d_in[3];
  const float* nu_r = (const float*)d_in[4];
  const float* th_r = (const float*)d_in[5];
  const float* gm_r = (const float*)d_in[6];
  const float* U_re = (const float*)d_in[7];
  const float* U_im = (const float*)d_in[8];
  const float* V_re = (const float*)d_in[9];
  const float* V_im = (const float*)d_in[10];
  const float* pw_re = (const float*)d_in[11];
  const float* pw_im = (const float*)d_in[12];
  const float* pb_re = (const float*)d_in[13];
  const float* pb_im = (const float*)d_in[14];
  const float* cw_r = (const float*)d_in[15];
  const float* cb_r = (const float*)d_in[16];
  const float* cw_i = (const float*)d_in[17];
  const float* cb_i = (const float*)d_in[18];
  const float* fw   = (const float*)d_in[19];
  const float* fb   = (const float*)d_in[20];
  const float* ln_g = (const float*)d_in[21];
  const float* ln_b = (const float*)d_in[22];
  float* out = (float*)d_out;

  // Workspace layout (floats). Total ~33.7M floats (~135 MB).
  float* ws = (float*)d_ws;
  float* TFRE = ws + 0;       float* TFIM = ws + 16384;
  float* TIRE = ws + 32768;   float* TIIM = ws + 49152;
  float* LAMSRE = ws + 65536; float* LAMSIM = ws + 69632; float* GAMS = ws + 73728;
  float* LAMRRE = ws + 77824; float* LAMRIM = ws + 78848; float* GAMR = ws + 79872;
  float* ZINRE = ws + 80896;  float* ZINIM = ws + 97280;
  float* ZSRE = ws + 113664;  float* ZSIM = ws + 130048;
  float* ARE = ws + 146432;            float* AIM = ARE + 8388608;
  float* BRE = AIM + 8388608;          float* BIM = BRE + 8388608;

  clru_init<<<dim3(64), 256, 0, stream>>>(
      TFRE, TFIM, TIRE, TIIM, nu_s, th_s, gm_s, nu_r, th_r, gm_r,
      LAMSRE, LAMSIM, GAMS, LAMRRE, LAMRIM, GAMR);

  // Forward row DFT (x real): B = x @ F
  clru_cgemm128<true><<<dim3(512), 256, 0, stream>>>(
      BRE, BIM, 16384, x, nullptr, 16384, TFRE, TFIM, 0);
  // Forward col DFT: A = F @ B  (A = xc)
  clru_cgemm128<false><<<dim3(512), 256, 0, stream>>>(
      ARE, AIM, 16384, TFRE, TFIM, 0, BRE, BIM, 16384);

  // Low-rank input projection + recurrences
  clru_zin<<<dim3(512), 256, 0, stream>>>(ARE, AIM, U_re, U_im, V_re, V_im,
                                          ZINRE, ZINIM);
  clru_zscan<<<dim3(4), 256, 0, stream>>>(ZINRE, ZINIM, LAMRRE, LAMRIM, GAMR,
                                          ZSRE, ZSIM);
  clru_hscan<<<dim3(2048), 256, 0, stream>>>(ARE, AIM, LAMSRE, LAMSIM, GAMS);
  clru_expand<<<dim3(512), 256, 0, stream>>>(ARE, AIM, U_re, U_im, V_re, V_im,
                                             ZSRE, ZSIM);

  // Channel mix: B = pw^T @ A + pb
  clru_chanmix<<<dim3(128, 16), 256, 0, stream>>>(ARE, AIM, pw_re, pw_im,
                                                  pb_re, pb_im, BRE, BIM);

  // Inverse col DFT: A = Finv @ B ; inverse row DFT: B = A @ Finv (= ys)
  clru_cgemm128<false><<<dim3(512), 256, 0, stream>>>(
      ARE, AIM, 16384, TIRE, TIIM, 0, BRE, BIM, 16384);
  clru_cgemm128<false><<<dim3(512), 256, 0, stream>>>(
      BRE, BIM, 16384, ARE, AIM, 16384, TIRE, TIIM, 0);

  // Convolutions (real / imag) -> A.re / A.im
  clru_conv3x3<<<dim3(4, 8, 16), 256, 0, stream>>>(BRE, cw_r, cb_r, ARE);
  clru_conv3x3<<<dim3(4, 8, 16), 256, 0, stream>>>(BIM, cw_i, cb_i, AIM);

  // Final projection -> B.re ; LayerNorm + residual -> out
  clru_proj<<<dim3(128, 16), 256, 0, stream>>>(ARE, AIM, fw, fb, BRE);
  clru_ln_res<<<dim3(512), 256, 0, stream>>>(BRE, x, ln_g, ln_b, out);
}